// GraphClassifier_56470230008285
// MI455X (gfx1250) — compile-verified
//
#include <hip/hip_runtime.h>

#define N_NODES 50000
#define N_EDGES 800000
#define D_IN    128
#define H_DIM   256
#define G_GRAPHS 128
#define O_OUT   10
#define BN_EPS  1e-5f
#define SLOPE   0.2f

// padded pair-row stride (floats): 160 mod 64 == 32 -> lane halves hit disjoint LDS banks
#define BPAIR_STRIDE 160

typedef __attribute__((ext_vector_type(2))) float v2f;
typedef __attribute__((ext_vector_type(8))) float v8f;

__global__ void zero_kernel(float* __restrict__ p, long n) {
    long i = (long)blockIdx.x * blockDim.x + threadIdx.x;
    if (i < n) p[i] = 0.0f;
}

// agg[dst] += relu(x[src] + edge_attr @ We + be), 4 features per thread
__global__ void edge_msg_kernel(const float* __restrict__ x,
                                const int*   __restrict__ src,
                                const int*   __restrict__ dst,
                                const float* __restrict__ ea,
                                const float* __restrict__ We,   // [3, din]
                                const float* __restrict__ be,   // [din]
                                float*       __restrict__ agg,  // [N, din]
                                int vshift, int din) {
    long tid   = (long)blockIdx.x * blockDim.x + threadIdx.x;
    long total = (long)N_EDGES << vshift;
    if (tid >= total) return;
    int e = (int)(tid >> vshift);
    int f = (int)(tid & ((1 << vshift) - 1)) << 2;
    int s = src[e], d = dst[e];
    float a0 = ea[e * 3 + 0], a1 = ea[e * 3 + 1], a2 = ea[e * 3 + 2];
    const float4 xv = *(const float4*)(x + (long)s * din + f);
    const float* w0 = We + f;
    const float* w1 = We + din + f;
    const float* w2 = We + 2 * din + f;
    float m0 = xv.x + fmaf(a0, w0[0], fmaf(a1, w1[0], fmaf(a2, w2[0], be[f + 0])));
    float m1 = xv.y + fmaf(a0, w0[1], fmaf(a1, w1[1], fmaf(a2, w2[1], be[f + 1])));
    float m2 = xv.z + fmaf(a0, w0[2], fmaf(a1, w1[2], fmaf(a2, w2[2], be[f + 2])));
    float m3 = xv.w + fmaf(a0, w0[3], fmaf(a1, w1[3], fmaf(a2, w2[3], be[f + 3])));
    m0 = fmaxf(m0, 0.0f); m1 = fmaxf(m1, 0.0f);
    m2 = fmaxf(m2, 0.0f); m3 = fmaxf(m3, 0.0f);
    float* ap = agg + (long)d * din + f;
    atomicAdd(ap + 0, m0);
    atomicAdd(ap + 1, m1);
    atomicAdd(ap + 2, m2);
    atomicAdd(ap + 3, m3);
}

// fp32 WMMA GEMM: out[M,256] = epilogue( (X (+AGG)) @ W[K,256] )
// Block: 4 waves, each owning a 16-row tile x 64 columns; the 64-column W panel
// is staged in LDS once per block, repacked so (k,k+1) pairs are contiguous
// per column -> inner loop does ds_load_b64 B fragments.
// EPI==0: BN(eval)+bias fold + LeakyReLU (p0=b1,p1=gamma,p2=beta,p3=mu,p4=var)
// EPI==1: bias + LeakyReLU               (p0=b2)
template <int EPI, int HAS_AGG>
__global__ void wmma_gemm_kernel(const float* __restrict__ X,
                                 const float* __restrict__ AGG,
                                 const float* __restrict__ W,    // [K, 256]
                                 float*       __restrict__ out,  // [M, 256]
                                 int M, int K,
                                 const float* __restrict__ p0,
                                 const float* __restrict__ p1,
                                 const float* __restrict__ p2,
                                 const float* __restrict__ p3,
                                 const float* __restrict__ p4) {
    extern __shared__ float lds[];   // (K/2) * BPAIR_STRIDE floats

    const int lane = threadIdx.x & 31;
    const int wave = blockIdx.x * (blockDim.x >> 5) + (threadIdx.x >> 5);
    const int mtiles = M >> 4;
    const int col0 = blockIdx.y << 6;           // 64 columns per block

    // --- stage W[0:K][col0:col0+64] into LDS, pair-packed:
    //     lds[(k>>1)*STRIDE + c*2 + (k&1)] = W[k][col0+c]
    for (int i = threadIdx.x; i < (K << 6); i += blockDim.x) {
        int k = i >> 6;
        int c = i & 63;
        lds[(k >> 1) * BPAIR_STRIDE + c * 2 + (k & 1)] = W[(long)k * H_DIM + col0 + c];
    }
    __syncthreads();

    if (wave < mtiles) {                        // wave-uniform: EXEC all-1s for WMMA
        const int row0 = wave << 4;
        const int mn   = lane & 15;
        const int khalf = lane >> 4;            // 0 or 1 -> kk = 2*khalf

        v8f acc0 = {}, acc1 = {}, acc2 = {}, acc3 = {};
        const float* arow = X + (long)(row0 + mn) * K + 2 * khalf;
        const float* grow = HAS_AGG ? AGG + (long)(row0 + mn) * K + 2 * khalf : nullptr;

        for (int k = 0; k < K; k += 4) {
            v2f a;
            if (HAS_AGG) {
                a.x = arow[k] + grow[k];
                a.y = arow[k + 1] + grow[k + 1];
            } else {
                a.x = arow[k];
                a.y = arow[k + 1];
            }
            const v2f* bp = (const v2f*)(lds + ((k >> 1) + khalf) * BPAIR_STRIDE + mn * 2);
            v2f b0 = bp[0];
            v2f b1 = bp[16];
            v2f b2 = bp[32];
            v2f b3 = bp[48];
            acc0 = __builtin_amdgcn_wmma_f32_16x16x4_f32(false, a, false, b0, (short)0, acc0, false, false);
            acc1 = __builtin_amdgcn_wmma_f32_16x16x4_f32(false, a, false, b1, (short)0, acc1, false, false);
            acc2 = __builtin_amdgcn_wmma_f32_16x16x4_f32(false, a, false, b2, (short)0, acc2, false, false);
            acc3 = __builtin_amdgcn_wmma_f32_16x16x4_f32(false, a, false, b3, (short)0, acc3, false, false);
        }

        const int mtop = khalf << 3;            // 0 or 8
        v8f accs[4] = {acc0, acc1, acc2, acc3};
#pragma unroll
        for (int t = 0; t < 4; ++t) {
            const int n = col0 + t * 16 + mn;
            float scale, shift;
            if (EPI == 0) {
                float s = p1[n] * rsqrtf(p4[n] + BN_EPS);
                scale = s;
                shift = p0[n] * s + p2[n] - p3[n] * s;  // fold bias + BN shift
            } else {
                scale = 1.0f;
                shift = p0[n];
            }
#pragma unroll
            for (int r = 0; r < 8; ++r) {
                float v = accs[t][r] * scale + shift;
                v = v > 0.0f ? v : SLOPE * v;           // LeakyReLU
                out[(long)(row0 + mtop + r) * H_DIM + n] = v;
            }
        }
    }
}

// global_add_pool: pool[batch[n]][off + f] += x[n][f]
__global__ void pool_kernel(const float* __restrict__ x,
                            const int*   __restrict__ batch,
                            float*       __restrict__ pool, int off) {
    long tid = (long)blockIdx.x * blockDim.x + threadIdx.x;
    if (tid >= (long)N_NODES * H_DIM) return;
    int n = (int)(tid >> 8);
    int f = (int)(tid & (H_DIM - 1));
    atomicAdd(&pool[(long)batch[n] * (3 * H_DIM) + off + f], x[tid]);
}

// small MLP layer on [G,K] -> [G,Nout]; act2=1 applies LeakyReLU twice
__global__ void mlp_kernel(const float* __restrict__ in,
                           const float* __restrict__ W,   // [K, Nout]
                           const float* __restrict__ b,
                           float*       __restrict__ out,
                           int K, int Nout, int act2) {
    int tid = blockIdx.x * blockDim.x + threadIdx.x;
    if (tid >= G_GRAPHS * Nout) return;
    int g = tid / Nout;
    int j = tid - g * Nout;
    const float* row = in + (long)g * K;
    float acc = b[j];
    for (int k = 0; k < K; ++k) acc = fmaf(row[k], W[(long)k * Nout + j], acc);
    if (act2) {
        acc = acc > 0.0f ? acc : SLOPE * acc;
        acc = acc > 0.0f ? acc : SLOPE * acc;
    }
    out[tid] = acc;
}

static inline long cdiv(long a, long b) { return (a + b - 1) / b; }

extern "C" void kernel_launch(void* const* d_in, const int* in_sizes, int n_in,
                              void* d_out, int out_size, void* d_ws, size_t ws_size,
                              hipStream_t stream) {
    const float* x     = (const float*)d_in[0];
    const int*   ei    = (const int*)d_in[1];
    const int*   src   = ei;
    const int*   dst   = ei + N_EDGES;
    const float* ea    = (const float*)d_in[2];
    const int*   batch = (const int*)d_in[3];
    // layer params: base 4 + 10*(layer-1); order We,be,W1,b1,g,bt,mu,var,W2,b2
    auto LP = [&](int layer, int j) { return (const float*)d_in[4 + 10 * (layer - 1) + j]; };
    const float* Wm[4] = {(const float*)d_in[34], (const float*)d_in[36],
                          (const float*)d_in[38], (const float*)d_in[40]};
    const float* bm[4] = {(const float*)d_in[35], (const float*)d_in[37],
                          (const float*)d_in[39], (const float*)d_in[41]};

    float* ws   = (float*)d_ws;
    float* agg  = ws;                                // N*256
    float* hmid = agg  + (long)N_NODES * H_DIM;      // N*256
    float* bufA = hmid + (long)N_NODES * H_DIM;      // N*256 (x1 / x3)
    float* bufB = bufA + (long)N_NODES * H_DIM;      // N*256 (x2)
    float* pool = bufB + (long)N_NODES * H_DIM;      // G*768
    float* m1   = pool + (long)G_GRAPHS * 3 * H_DIM; // G*200
    float* m2   = m1   + (long)G_GRAPHS * 200;       // G*200

    // zero pooled sums once per call
    {
        long n = (long)G_GRAPHS * 3 * H_DIM;
        zero_kernel<<<dim3((unsigned)cdiv(n, 256)), dim3(256), 0, stream>>>(pool, n);
    }

    const int mtiles = N_NODES / 16;                 // 3125
    dim3 gemm_grid((unsigned)cdiv(mtiles, 4), H_DIM / 64);
    dim3 gemm_block(128);

    for (int layer = 1; layer <= 3; ++layer) {
        const int din    = (layer == 1) ? D_IN : H_DIM;
        const int vshift = (din == D_IN) ? 5 : 6;    // din/4 threads per edge
        const float* xin = (layer == 1) ? x : (layer == 2 ? bufA : bufB);
        float* xout      = (layer == 1) ? bufA : (layer == 2 ? bufB : bufA);
        const size_t lds1 = (size_t)(din / 2) * BPAIR_STRIDE * sizeof(float);
        const size_t lds2 = (size_t)(H_DIM / 2) * BPAIR_STRIDE * sizeof(float);

        // agg = 0
        {
            long n = (long)N_NODES * din;
            zero_kernel<<<dim3((unsigned)cdiv(n, 256)), dim3(256), 0, stream>>>(agg, n);
        }
        // scatter messages
        {
            long total = (long)N_EDGES << vshift;
            edge_msg_kernel<<<dim3((unsigned)cdiv(total, 256)), dim3(256), 0, stream>>>(
                xin, src, dst, ea, LP(layer, 0), LP(layer, 1), agg, vshift, din);
        }
        // hmid = leaky(BN((xin+agg) @ W1 + b1))
        wmma_gemm_kernel<0, 1><<<gemm_grid, gemm_block, lds1, stream>>>(
            xin, agg, LP(layer, 2), hmid, N_NODES, din,
            LP(layer, 3), LP(layer, 4), LP(layer, 5), LP(layer, 6), LP(layer, 7));
        // xout = leaky(hmid @ W2 + b2)
        wmma_gemm_kernel<1, 0><<<gemm_grid, gemm_block, lds2, stream>>>(
            hmid, nullptr, LP(layer, 8), xout, N_NODES, H_DIM,
            LP(layer, 9), nullptr, nullptr, nullptr, nullptr);
        // pool xout into segment (layer-1)*H
        {
            long n = (long)N_NODES * H_DIM;
            pool_kernel<<<dim3((unsigned)cdiv(n, 256)), dim3(256), 0, stream>>>(
                xout, batch, pool, (layer - 1) * H_DIM);
        }
    }

    // MLP head: 768 -> 200 -> 200 -> 200 -> 10, doubled LeakyReLU on hidden layers
    mlp_kernel<<<dim3((unsigned)cdiv((long)G_GRAPHS * 200, 128)), dim3(128), 0, stream>>>(
        pool, Wm[0], bm[0], m1, 3 * H_DIM, 200, 1);
    mlp_kernel<<<dim3((unsigned)cdiv((long)G_GRAPHS * 200, 128)), dim3(128), 0, stream>>>(
        m1, Wm[1], bm[1], m2, 200, 200, 1);
    mlp_kernel<<<dim3((unsigned)cdiv((long)G_GRAPHS * 200, 128)), dim3(128), 0, stream>>>(
        m2, Wm[2], bm[2], m1, 200, 200, 1);
    mlp_kernel<<<dim3((unsigned)cdiv((long)G_GRAPHS * O_OUT, 128)), dim3(128), 0, stream>>>(
        m1, Wm[3], bm[3], (float*)d_out, 200, O_OUT, 0);
}